// NSM_19988777796013
// MI455X (gfx1250) — compile-verified
//
#include <hip/hip_runtime.h>
#include <hip/hip_bf16.h>
#include <math.h>

// ---------------------------------------------------------------------------
// NSM pipeline for MI455X (gfx1250, wave32).
//  * fp32 WMMA (V_WMMA_F32_16X16X4_F32) for all GEMMs (outputs feed argsorts,
//    so fp32 precision is required).
//  * trans_e = edge_attr @ W_edge (192 MB) and trans = node_attr @ W_props are
//    timestep-invariant -> computed ONCE (reference recomputes them T=5 times).
//    trans_e (192 MB) fits the 192 MB L2 -> 5 streaming replay passes are
//    L2-resident. Workload is bandwidth bound (~1.5 GB -> ~65us at 23.3 TB/s).
//  * W-strip (K x 16) staged into LDS per block via the Tensor Data Mover
//    (tensor_load_to_lds + s_wait_tensorcnt), shared by 4 waves (64x16 tile).
//  * Workspace requirement: ~203 MB (50,734,528 floats).
// ---------------------------------------------------------------------------

#define Hc 300
#define Pc 4
#define Tc 5
#define Bc 16
#define Nc 100
#define Cc 500
#define Dc 5   // P+1

#ifndef NSM_USE_TDM
#define NSM_USE_TDM 1
#endif

typedef float v2f __attribute__((ext_vector_type(2)));
typedef float v8f __attribute__((ext_vector_type(8)));
typedef unsigned int u32x4 __attribute__((ext_vector_type(4)));
typedef int i32x4 __attribute__((ext_vector_type(4)));
typedef int i32x8 __attribute__((ext_vector_type(8)));

__device__ __forceinline__ float eluf(float x) {
    return x > 0.0f ? x : (expf(x) - 1.0f);
}

// ---------------------------------------------------------------------------
// TDM: stage a (K rows x 16 cols) fp32 strip of row-major W (ld = ldw elems)
// starting at column col0 into LDS at lds_off. validw = Ncols - col0 gives
// hardware OOB zero-fill for the ragged last tile. D# per CDNA5 ISA 8.3-8.6.
// ---------------------------------------------------------------------------
#if NSM_USE_TDM
__device__ __forceinline__ void tdm_load_strip(const float* gsrc, unsigned lds_off,
                                               int validw, int K, int ldw)
{
    const unsigned long long ga = (unsigned long long)(size_t)gsrc;
    u32x4 g0;
    g0.x = 1u;                                                  // count=1, user mode
    g0.y = lds_off;                                             // lds_addr (bytes)
    g0.z = (unsigned)ga;                                        // global_addr[31:0]
    g0.w = (unsigned)((ga >> 32) & 0x01FFFFFFu) | (2u << 30);   // addr[56:32] | type=2
    if (validw > 16) validw = 16;
    i32x8 g1;
    g1[0] = 2 << 16;                                            // data_size=4B; no mask/pad
    g1[1] = (validw & 0xFFFF) << 16;                            // tensor_dim0[15:0]
    g1[2] = ((unsigned)validw >> 16) | ((K & 0xFFFF) << 16);    // td0[31:16] | tensor_dim1 lo
    g1[3] = ((unsigned)K >> 16) | (16 << 16);                   // td1 hi | tile_dim0=16
    g1[4] = (K & 0xFFFF);                                       // tile_dim1=K, tile_dim2=0
    g1[5] = ldw;                                                // tensor_dim0_stride[31:0]
    g1[6] = 0;                                                  // stride0 hi | stride1 lo
    g1[7] = 0;
    i32x4 z4 = {0, 0, 0, 0};
#if __has_include(<hip/amd_detail/amd_gfx1250_TDM.h>)
    i32x8 z8 = {0, 0, 0, 0, 0, 0, 0, 0};
    __builtin_amdgcn_tensor_load_to_lds(g0, g1, z4, z4, z8, 0);   // 6-arg toolchain
#else
    __builtin_amdgcn_tensor_load_to_lds(g0, g1, z4, z4, 0);       // 5-arg toolchain
#endif
    __builtin_amdgcn_s_wait_tensorcnt(0);
}
#endif

// ---------------------------------------------------------------------------
// Specialized GEMM for C = A @ W, W row-major [K x Ncols] (ldbk=ldw, ldbn=1).
// Block = 128 threads = 4 waves -> 64x16 output tile; one W strip (K x 16)
// staged in LDS (TDM) and shared by the 4 waves. Requires M % 64 == 0.
// ---------------------------------------------------------------------------
__global__ __launch_bounds__(128)
void nsm_wmma_gemm_w(const float* __restrict__ A, int lda,
                     const float* __restrict__ W, int ldw,
                     float* __restrict__ C, int ldc,
                     int M, int Ncols, int K)
{
    __shared__ float sB[Hc * 16];            // K x 16 strip, row-major (k*16 + c)
    const int tid  = threadIdx.x;
    const int lane = tid & 31;
    const int wav  = tid >> 5;
    const int tn   = blockIdx.x;
    const int col0 = tn * 16;

#if NSM_USE_TDM
    tdm_load_strip(W + col0, (unsigned)(size_t)sB, Ncols - col0, K, ldw);
    __syncthreads();
#else
    for (int i = tid; i < K * 16; i += 128) {
        const int k = i >> 4, c = i & 15;
        sB[i] = (col0 + c < Ncols) ? W[(size_t)k * ldw + col0 + c] : 0.0f;
    }
    __syncthreads();
#endif

    const int half = lane >> 4;              // 0 or 1
    const int l15  = lane & 15;
    const int row  = (blockIdx.y * 4 + wav) * 16 + l15;
    const float* __restrict__ Arow = A + (size_t)row * lda;

    v8f acc = {};
#pragma unroll 5
    for (int k0 = 0; k0 < K; k0 += 4) {
        const int kb = k0 + half * 2;
        v2f a, b;
        a.x = Arow[kb];
        a.y = Arow[kb + 1];
        b.x = sB[kb * 16 + l15];
        b.y = sB[(kb + 1) * 16 + l15];
        acc = __builtin_amdgcn_wmma_f32_16x16x4_f32(false, a, false, b,
                                                    (short)0, acc, false, false);
    }

    const int c = col0 + l15;
    if (c < Ncols) {
        const int rbase = (blockIdx.y * 4 + wav) * 16 + half * 8;
#pragma unroll
        for (int i = 0; i < 8; ++i)
            C[(size_t)(rbase + i) * ldc + c] = acc[i];
    }
}

// ---------------------------------------------------------------------------
// Generic fp32 WMMA GEMM: C[M x Ncols] = A[M x K] @ B, B element (k,n) at
// Bm[k*ldbk + n*ldbn]. One wave per 16x16 tile. Branch-free OOB handling
// (clamped index + 0/1 multiplier) -> no EXEC churn in the K loop.
// ---------------------------------------------------------------------------
__global__ __launch_bounds__(32)
void nsm_wmma_gemm_f32(const float* __restrict__ A, int lda,
                       const float* __restrict__ Bm, int ldbk, int ldbn,
                       float* __restrict__ C, int ldc,
                       int M, int Ncols, int K)
{
    const int lane = threadIdx.x;
    const int tn   = blockIdx.x;
    const int tm   = blockIdx.y;
    const int half = lane >> 4;
    const int l15  = lane & 15;
    const int row  = tm * 16 + l15;
    const int col  = tn * 16 + l15;
    const bool colOK = (col < Ncols);
    const int colC   = colOK ? col : (Ncols - 1);   // always-legal address
    const float bsel = colOK ? 1.0f : 0.0f;         // zero contribution if OOB

    const float* __restrict__ Arow = A + (size_t)row * lda;
    const float* __restrict__ Bcol = Bm + (size_t)colC * ldbn;

    v8f acc = {};
#pragma unroll 5
    for (int k0 = 0; k0 < K; k0 += 4) {
        const int kb = k0 + half * 2;
        v2f a, b;
        a.x = Arow[kb];
        a.y = Arow[kb + 1];
        b.x = Bcol[(size_t)kb * ldbk] * bsel;
        b.y = Bcol[(size_t)(kb + 1) * ldbk] * bsel;
        acc = __builtin_amdgcn_wmma_f32_16x16x4_f32(false, a, false, b,
                                                    (short)0, acc, false, false);
    }

#pragma unroll
    for (int i = 0; i < 8; ++i) {
        const int r = tm * 16 + i + half * 8;
        const int c = tn * 16 + l15;
        if (r < M && c < Ncols) C[(size_t)r * ldc + c] = acc[i];
    }
}

// prop_keys[d][k] = sum_h property_embeddings[d][h] * W_p[h][k]   (tiny)
__global__ void nsm_prop_keys(const float* __restrict__ pe,
                              const float* __restrict__ Wp,
                              float* __restrict__ pk)
{
    int i = blockIdx.x * blockDim.x + threadIdx.x;
    if (i >= Dc * Hc) return;
    int d = i / Hc, k = i - d * Hc;
    float acc = 0.0f;
    for (int h = 0; h < Hc; ++h) acc += pe[d * Hc + h] * Wp[h * Hc + k];
    pk[i] = acc;
}

// dist0 = softmax(1/context_size[b] + node_mask[b,:]) over N
__global__ __launch_bounds__(128)
void nsm_dist0(const float* __restrict__ ctx, const float* __restrict__ mask,
               float* __restrict__ dist)
{
    const int b = blockIdx.x, tid = threadIdx.x;
    __shared__ float sh[128];
    const float inv = 1.0f / ctx[b];
    float lg = (tid < Nc) ? inv + mask[b * Nc + tid] : -INFINITY;
    sh[tid] = lg; __syncthreads();
    for (int s = 64; s > 0; s >>= 1) { if (tid < s) sh[tid] = fmaxf(sh[tid], sh[tid + s]); __syncthreads(); }
    const float mx = sh[0]; __syncthreads();
    float e = (tid < Nc) ? expf(lg - mx) : 0.0f;
    sh[tid] = e; __syncthreads();
    for (int s = 64; s > 0; s >>= 1) { if (tid < s) sh[tid] += sh[tid + s]; __syncthreads(); }
    if (tid < Nc) dist[b * Nc + tid] = e / sh[0];
}

// instr_prop = softmax_d(instr_t @ prop_keys^T); write ins_simi + ws psim
__global__ __launch_bounds__(32)
void nsm_instr_prop(const float* __restrict__ instr_all, int t,
                    const float* __restrict__ pk,
                    float* __restrict__ psim, float* __restrict__ out_simi)
{
    const int b = blockIdx.x, tid = threadIdx.x;
    __shared__ float lg[Dc];
    if (tid < Dc) {
        const float* ins = instr_all + ((size_t)b * Tc + t) * Hc;
        float acc = 0.0f;
        for (int h = 0; h < Hc; ++h) acc += ins[h] * pk[tid * Hc + h];
        lg[tid] = acc;
    }
    __syncthreads();
    if (tid == 0) {
        float mx = lg[0];
        for (int d = 1; d < Dc; ++d) mx = fmaxf(mx, lg[d]);
        float e[Dc], s = 0.0f;
        for (int d = 0; d < Dc; ++d) { e[d] = expf(lg[d] - mx); s += e[d]; }
        for (int d = 0; d < Dc; ++d) {
            const float v = e[d] / s;
            psim[b * Dc + d] = v;
            out_simi[((size_t)b * Tc + t) * Dc + d] = v;
        }
    }
}

// states[b,n] = softmax_n( sum_h w_state[h]*elu(instr[h]*sum_p psim[p]*trans[p,bn,h]) + mask )
__global__ __launch_bounds__(128)
void nsm_states(const float* __restrict__ tN, const float* __restrict__ instr_all, int t,
                const float* __restrict__ psim, const float* __restrict__ w_state,
                const float* __restrict__ mask, float* __restrict__ states)
{
    const int b = blockIdx.x, tid = threadIdx.x;
    __shared__ float si[Hc], sw[Hc], sh[128];
    for (int h = tid; h < Hc; h += 128) {
        si[h] = instr_all[((size_t)b * Tc + t) * Hc + h];
        sw[h] = w_state[h];
    }
    __syncthreads();
    const float p0 = psim[b * Dc + 0], p1 = psim[b * Dc + 1];
    const float p2 = psim[b * Dc + 2], p3 = psim[b * Dc + 3];
    float lg = -INFINITY;
    if (tid < Nc) {
        const size_t base = ((size_t)b * Nc + tid) * Hc;
        const size_t ps   = (size_t)Bc * Nc * Hc;
        const float* t0 = tN + base;
        const float* t1 = tN + ps + base;
        const float* t2 = tN + 2 * ps + base;
        const float* t3 = tN + 3 * ps + base;
        float acc = 0.0f;
        for (int h = 0; h < Hc; ++h) {
            const float s = p0 * t0[h] + p1 * t1[h] + p2 * t2[h] + p3 * t3[h];
            acc += sw[h] * eluf(si[h] * s);
        }
        lg = acc + mask[b * Nc + tid];
    }
    sh[tid] = lg; __syncthreads();
    for (int s = 64; s > 0; s >>= 1) { if (tid < s) sh[tid] = fmaxf(sh[tid], sh[tid + s]); __syncthreads(); }
    const float mx = sh[0]; __syncthreads();
    float e = (tid < Nc) ? expf(lg - mx) : 0.0f;
    sh[tid] = e; __syncthreads();
    for (int s = 64; s > 0; s >>= 1) { if (tid < s) sh[tid] += sh[tid + s]; __syncthreads(); }
    if (tid < Nc) states[b * Nc + tid] = e / sh[0];
}

// rel_logit[b,i] = sum_{j,h} dist[b,j]*w_rel[h]*elu(instr[h]*trans_e[b,i,j,h])
// Streams the L2-resident 192MB trans_e; one block per (b,i) row (120KB each).
__global__ __launch_bounds__(256)
void nsm_rel_logits(const float* __restrict__ tE, const float* __restrict__ instr_all, int t,
                    const float* __restrict__ dist, const float* __restrict__ w_rel,
                    float* __restrict__ rlog)
{
    const int bi = blockIdx.x;           // b*N + i
    const int b  = bi / Nc;
    const int tid = threadIdx.x;
    __shared__ float si[Hc], sw[Hc], sd[Nc], sh[256];
    for (int h = tid; h < Hc; h += 256) {
        si[h] = instr_all[((size_t)b * Tc + t) * Hc + h];
        sw[h] = w_rel[h];
    }
    for (int j = tid; j < Nc; j += 256) sd[j] = dist[b * Nc + j];
    __syncthreads();
    const float* __restrict__ rowbase = tE + (size_t)bi * Nc * Hc;
    float acc = 0.0f;
    int j = tid / Hc, h = tid - j * Hc;            // incremental (j,h); no div in loop
    for (int idx = tid; idx < Nc * Hc; idx += 256) {
        __builtin_prefetch(&rowbase[idx + 256 * 16], 0, 0); // global_prefetch ahead of stream
        acc += sd[j] * sw[h] * eluf(si[h] * rowbase[idx]);
        h += 256;
        if (h >= Hc) { h -= Hc; ++j; }
    }
    sh[tid] = acc; __syncthreads();
    for (int s = 128; s > 0; s >>= 1) { if (tid < s) sh[tid] += sh[tid + s]; __syncthreads(); }
    if (tid == 0) rlog[bi] = sh[0];
}

// relations = softmax(rlog + mask); dist = rel_sim*relations + (1-rel_sim)*states
__global__ __launch_bounds__(128)
void nsm_dist_update(const float* __restrict__ rlog, const float* __restrict__ mask,
                     const float* __restrict__ states, const float* __restrict__ psim,
                     float* __restrict__ dist)
{
    const int b = blockIdx.x, tid = threadIdx.x;
    __shared__ float sh[128];
    float lg = (tid < Nc) ? rlog[b * Nc + tid] + mask[b * Nc + tid] : -INFINITY;
    sh[tid] = lg; __syncthreads();
    for (int s = 64; s > 0; s >>= 1) { if (tid < s) sh[tid] = fmaxf(sh[tid], sh[tid + s]); __syncthreads(); }
    const float mx = sh[0]; __syncthreads();
    float e = (tid < Nc) ? expf(lg - mx) : 0.0f;
    sh[tid] = e; __syncthreads();
    for (int s = 64; s > 0; s >>= 1) { if (tid < s) sh[tid] += sh[tid + s]; __syncthreads(); }
    if (tid < Nc) {
        const float rel = e / sh[0];
        const float rs  = psim[b * Dc + (Dc - 1)];
        dist[b * Nc + tid] = rs * rel + (1.0f - rs) * states[b * Nc + tid];
    }
}

// Top-20 of 500 per row, descending, stable ties (smallest index) == argsort(-x)[:20]
__global__ __launch_bounds__(128)
void nsm_top20(const float* __restrict__ s0, float* __restrict__ data0, float* __restrict__ idx0)
{
    const int row = blockIdx.x, tid = threadIdx.x;
    __shared__ float vals[Cc];
    __shared__ float rv[128];
    __shared__ int   ri[128];
    for (int c = tid; c < Cc; c += 128) vals[c] = s0[(size_t)row * Cc + c];
    __syncthreads();
    for (int k = 0; k < 20; ++k) {
        float bv = -INFINITY; int bi = Cc;
        for (int c = tid; c < Cc; c += 128) {
            const float v = vals[c];
            if (v > bv || (v == bv && c < bi)) { bv = v; bi = c; }
        }
        rv[tid] = bv; ri[tid] = bi; __syncthreads();
        for (int s = 64; s > 0; s >>= 1) {
            if (tid < s) {
                const float v2 = rv[tid + s]; const int i2 = ri[tid + s];
                if (v2 > rv[tid] || (v2 == rv[tid] && i2 < ri[tid])) { rv[tid] = v2; ri[tid] = i2; }
            }
            __syncthreads();
        }
        if (tid == 0) {
            data0[(size_t)row * 20 + k] = rv[0];
            idx0[(size_t)row * 20 + k]  = (float)ri[0];
            vals[ri[0]] = -INFINITY;
        }
        __syncthreads();
    }
}

// Top-100 of 500 per (b); values mapped through softmax over ALL 500 entries.
__global__ __launch_bounds__(256)
void nsm_top100(const float* __restrict__ sbuf, int t,
                float* __restrict__ insdata, float* __restrict__ insidx)
{
    const int b = blockIdx.x, tid = threadIdx.x;
    __shared__ float vals[Cc];
    __shared__ float rv[256];
    __shared__ int   ri[256];
    for (int c = tid; c < Cc; c += 256) vals[c] = sbuf[(size_t)b * Cc + c];
    __syncthreads();
    float m = -INFINITY;
    for (int c = tid; c < Cc; c += 256) m = fmaxf(m, vals[c]);
    rv[tid] = m; __syncthreads();
    for (int s = 128; s > 0; s >>= 1) { if (tid < s) rv[tid] = fmaxf(rv[tid], rv[tid + s]); __syncthreads(); }
    const float mx = rv[0]; __syncthreads();
    float ps = 0.0f;
    for (int c = tid; c < Cc; c += 256) ps += expf(vals[c] - mx);
    rv[tid] = ps; __syncthreads();
    for (int s = 128; s > 0; s >>= 1) { if (tid < s) rv[tid] += rv[tid + s]; __syncthreads(); }
    const float denom = rv[0]; __syncthreads();
    for (int k = 0; k < 100; ++k) {
        float bv = -INFINITY; int bi = Cc;
        for (int c = tid; c < Cc; c += 256) {
            const float v = vals[c];
            if (v > bv || (v == bv && c < bi)) { bv = v; bi = c; }
        }
        rv[tid] = bv; ri[tid] = bi; __syncthreads();
        for (int s = 128; s > 0; s >>= 1) {
            if (tid < s) {
                const float v2 = rv[tid + s]; const int i2 = ri[tid + s];
                if (v2 > rv[tid] || (v2 == rv[tid] && i2 < ri[tid])) { rv[tid] = v2; ri[tid] = i2; }
            }
            __syncthreads();
        }
        if (tid == 0) {
            const size_t o = ((size_t)b * Tc + t) * 100 + k;
            insdata[o] = expf(rv[0] - mx) / denom;
            insidx[o]  = (float)ri[0];
            vals[ri[0]] = -INFINITY;
        }
        __syncthreads();
    }
}

__global__ void nsm_copy(const float* __restrict__ src, float* __restrict__ dst, int n)
{
    const int i = blockIdx.x * blockDim.x + threadIdx.x;
    if (i < n) dst[i] = src[i];
}

// ---------------------------------------------------------------------------
extern "C" void kernel_launch(void* const* d_in, const int* in_sizes, int n_in,
                              void* d_out, int out_size, void* d_ws, size_t ws_size,
                              hipStream_t stream)
{
    (void)in_sizes; (void)n_in; (void)out_size; (void)ws_size;

    const float* node_attr  = (const float*)d_in[0];   // (B,N,P,H)
    const float* edge_attr  = (const float*)d_in[1];   // (B,N,N,H)
    const float* instr_all  = (const float*)d_in[2];   // (B,T,H)
    const float* prop_emb   = (const float*)d_in[3];   // (D,H)
    const float* vocab      = (const float*)d_in[4];   // (C,H)
    const float* node_mask  = (const float*)d_in[5];   // (B,N)
    const float* ctx        = (const float*)d_in[6];   // (B,)
    const float* W_p        = (const float*)d_in[7];   // (H,H)
    const float* W_props    = (const float*)d_in[8];   // (P,H,H)
    const float* W_edge     = (const float*)d_in[9];   // (H,H)
    const float* w_state    = (const float*)d_in[10];  // (H,)
    const float* w_rel      = (const float*)d_in[11];  // (H,)

    float* out = (float*)d_out;
    float* ws  = (float*)d_ws;

    // ---- workspace layout (floats). Total 50,734,528 floats ~= 203 MB ----
    float* tE     = ws;                     // 48,000,000  trans_e (B*N*N, H)
    float* tN     = ws + 48000000;          //  1,920,000  trans   (P, B*N, H)
    float* dist   = ws + 49920000;          //      1,600
    float* pkeys  = ws + 49921600;          //      1,500 (padded 1,600)
    float* sbuf   = ws + 49923200;          //      8,000  (B,C) per-step scores
    float* states = ws + 49931200;          //      1,600
    float* rlog   = ws + 49932800;          //      1,600
    float* psim   = ws + 49934400;          //         80 (padded 128)
    float* s0buf  = ws + 49934528;          //    800,000  simi0 (B*N, C)

    // ---- output layout (floats, concatenated in return order) ----
    float* o_dist    = out;                 //  1,600
    float* o_data0   = out + 1600;          // 32,000
    float* o_idx0    = out + 33600;         // 32,000
    float* o_insdata = out + 65600;         //  8,000
    float* o_insidx  = out + 73600;         //  8,000
    float* o_simi    = out + 81600;         //    400

    nsm_prop_keys<<<(Dc * Hc + 255) / 256, 256, 0, stream>>>(prop_emb, W_p, pkeys);
    nsm_dist0<<<Bc, 128, 0, stream>>>(ctx, node_mask, dist);

    // trans_e = edge_attr @ W_edge (once)  M=160000, N=300, K=300. 64x16/block.
    {
        dim3 g((Hc + 15) / 16, (Bc * Nc * Nc) / 64);
        nsm_wmma_gemm_w<<<g, 128, 0, stream>>>(edge_attr, Hc, W_edge, Hc,
                                               tE, Hc, Bc * Nc * Nc, Hc, Hc);
    }
    // trans[p] = node_attr[:,:,p,:] @ W_props[p] (once)  M=1600, N=300, K=300
    for (int p = 0; p < Pc; ++p) {
        dim3 g((Hc + 15) / 16, (Bc * Nc) / 64);
        nsm_wmma_gemm_w<<<g, 128, 0, stream>>>(node_attr + p * Hc, Pc * Hc,
                                               W_props + (size_t)p * Hc * Hc, Hc,
                                               tN + (size_t)p * Bc * Nc * Hc, Hc,
                                               Bc * Nc, Hc, Hc);
    }
    // simi0 = node_attr[:,:,0,:] @ vocab^T   M=1600, N=500, K=300
    {
        dim3 g((Cc + 15) / 16, (Bc * Nc) / 16);
        nsm_wmma_gemm_f32<<<g, 32, 0, stream>>>(node_attr, Pc * Hc, vocab, 1, Hc,
                                                s0buf, Cc, Bc * Nc, Cc, Hc);
    }
    nsm_top20<<<Bc * Nc, 128, 0, stream>>>(s0buf, o_data0, o_idx0);

    // sequential NSM steps
    for (int t = 0; t < Tc; ++t) {
        nsm_instr_prop<<<Bc, 32, 0, stream>>>(instr_all, t, pkeys, psim, o_simi);
        nsm_states<<<Bc, 128, 0, stream>>>(tN, instr_all, t, psim, w_state, node_mask, states);
        nsm_rel_logits<<<Bc * Nc, 256, 0, stream>>>(tE, instr_all, t, dist, w_rel, rlog);
        nsm_dist_update<<<Bc, 128, 0, stream>>>(rlog, node_mask, states, psim, dist);
        // s = instr_t @ vocab^T : M=16 (one WMMA tile row), N=500, K=300
        {
            dim3 g((Cc + 15) / 16, 1);
            nsm_wmma_gemm_f32<<<g, 32, 0, stream>>>(instr_all + (size_t)t * Hc, Tc * Hc,
                                                    vocab, 1, Hc, sbuf, Cc, Bc, Cc, Hc);
        }
        nsm_top100<<<Bc, 256, 0, stream>>>(sbuf, t, o_insdata, o_insidx);
    }

    nsm_copy<<<(Bc * Nc + 255) / 256, 256, 0, stream>>>(dist, o_dist, Bc * Nc);
}